// ACSF_73461120631075
// MI455X (gfx1250) — compile-verified
//
#include <hip/hip_runtime.h>
#include <stdint.h>
#include <math.h>

// ---------------- problem constants (match reference setup_inputs) ----------
#define N_COL   70          // 16 G2 cols + 54 G4 cols per atom
#define NS      2           // number of species
#define TILE    256         // triplets per block-tile (8 wave32)
#define MAXBLK  3072        // persistent-grid cap for the G4 pipeline
#define RCUT    6.0f
#define PI_RC   0.52359877559829887f   // pi / 6.0

typedef unsigned int uint32x4 __attribute__((ext_vector_type(4)));
typedef int          int32x4  __attribute__((ext_vector_type(4)));
typedef int          int32x8  __attribute__((ext_vector_type(8)));

// ------------- CDNA5 Tensor-Data-Mover availability probe -------------------
#if defined(__HIP_DEVICE_COMPILE__) && defined(__gfx1250__) && \
    __has_builtin(__builtin_amdgcn_tensor_load_to_lds) && \
    __has_builtin(__builtin_amdgcn_s_wait_tensorcnt)
#define USE_TDM 1
#else
#define USE_TDM 0
#endif

__device__ __forceinline__ void atomic_add_f32(float* p, float v) {
  // hardware global_atomic_add_f32; accumulator (14MB) is L2-resident (192MB L2)
  unsafeAtomicAdd(p, v);
}

#if USE_TDM
__device__ __forceinline__ unsigned lds_off(const void* p) {
  // generic shared pointer: low 32 bits == wave-relative LDS byte offset
  return (unsigned)(uintptr_t)p;
}

// 1D contiguous DMA: nelem 32-bit words, global -> LDS.  Descriptor per
// CDNA5 ISA (count=1, type=2,
// data_size=4B, degenerate 2D tile: tile_dim0=nelem, tile_dim1=1).
__device__ __forceinline__ void tdm_load_1d_b32(const void* gsrc,
                                                unsigned lds_byte_off,
                                                int nelem) {
  unsigned long long ga = (unsigned long long)(uintptr_t)gsrc;
  uint32x4 g0;
  g0[0] = 1u;                                              // count=1 (user D#)
  g0[1] = lds_byte_off;                                    // lds_addr
  g0[2] = (unsigned)(ga & 0xFFFFFFFFull);                  // global_addr[31:0]
  g0[3] = (unsigned)((ga >> 32) & 0x1FFFFFFull)            // global_addr[56:32]
        | (2u << 30);                                      // type = 2 ("image")
  int32x8 g1;
  g1[0] = 0x00020000;                                      // data_size=2 (4B)
  g1[1] = (nelem & 0xFFFF) << 16;                          // tensor_dim0[15:0]
  g1[2] = ((nelem >> 16) & 0xFFFF) | (1 << 16);            // dim0[31:16] | tensor_dim1=1
  g1[3] = (nelem & 0xFFFF) << 16;                          // dim1 hi=0 | tile_dim0
  g1[4] = 1;                                               // tile_dim1=1, tile_dim2=0
  g1[5] = nelem;                                           // tensor_dim0_stride[31:0]
  g1[6] = (nelem & 0xFFFF) << 16;                          // stride0 hi=0 | stride1[15:0]
  g1[7] = 0;
  int32x4 z4 = {0, 0, 0, 0};                               // groups 2/3 unused (<=2D)
#if defined(__clang_major__) && __clang_major__ >= 23
  int32x8 z8 = {0, 0, 0, 0, 0, 0, 0, 0};
  __builtin_amdgcn_tensor_load_to_lds(g0, g1, z4, z4, z8, 0);
#else
  __builtin_amdgcn_tensor_load_to_lds(g0, g1, z4, z4, 0);
#endif
}

__device__ __forceinline__ void tdm_issue_tile(const int* id3_ba,
                                               const int* id3_ca,
                                               const float* cosphi,
                                               int tile, int T,
                                               int* sba, int* sca, float* scp) {
  int base = tile * TILE;
  int n = T - base; if (n > TILE) n = TILE;
  tdm_load_1d_b32(id3_ba + base, lds_off(sba), n);
  tdm_load_1d_b32(id3_ca + base, lds_off(sca), n);
  tdm_load_1d_b32(cosphi + base, lds_off(scp), n);
}
#endif

// ---------------------------------------------------------------------------
__global__ void acsf_zero(float* __restrict__ out, int n) {
  int i4 = (blockIdx.x * blockDim.x + threadIdx.x) * 4;
  if (i4 + 3 < n) {
    *(float4*)(out + i4) = make_float4(0.f, 0.f, 0.f, 0.f);
  } else {
    for (int k = i4; k < n; ++k) out[k] = 0.f;
  }
}

// One thread per edge: G2 scatter + per-edge {D, fc(D)*(D<rc)} table for G4.
__global__ void acsf_g2_edges(const int* __restrict__ ei,
                              const float* __restrict__ D_st,
                              const int* __restrict__ an,
                              const float* __restrict__ G2p,
                              const int* __restrict__ a2i,
                              const int* __restrict__ imap2,
                              float2* __restrict__ etab,
                              float* __restrict__ out,
                              int E, int maxa) {
  int e = blockIdx.x * blockDim.x + threadIdx.x;
  if (e >= E) return;
  float D   = D_st[e];
  float fcr = 0.5f * __cosf(PI_RC * D) + 0.5f;     // G2 uses UNMASKED fc
  if (etab) etab[e] = make_float2(D, (D < RCUT) ? fcr : 0.0f);
  int s = ei[e], d = ei[E + e];
  int as_ = an[s], ad = an[d];
  const float* eta = G2p + (a2i[ad] * NS + a2i[as_]) * 8;
  int desc2 = imap2[ad * maxa + as_];
  float D2 = D * D;
  float* base = out + (size_t)d * N_COL + desc2;    // col = k*NS + desc2
#pragma unroll
  for (int k = 0; k < 8; ++k)
    atomic_add_f32(base + 2 * k, fcr * __expf(-eta[k] * D2));
}

// Per-triplet G4 body (predicated; no early returns so it can sit in a
// barriered pipeline loop).
__device__ __forceinline__ void
g4_body(int ba, int ca, float cp,
        const int* __restrict__ ei, const float* __restrict__ D_st,
        const int* __restrict__ an, const float* __restrict__ G4e,
        const float* __restrict__ G4z, const float* __restrict__ G4l,
        const int* __restrict__ a2i, const int* __restrict__ imap,
        const float2* __restrict__ etab, float* __restrict__ out,
        int E, int maxa) {
  float D_ba, fba, D_ca, fca;
  if (etab) {
    float2 b = etab[ba]; D_ba = b.x; fba = b.y;      // fc pre-masked by D<rc
    float2 c = etab[ca]; D_ca = c.x; fca = c.y;
  } else {
    D_ba = D_st[ba]; D_ca = D_st[ca];
    fba = (D_ba < RCUT) ? 0.5f * __cosf(PI_RC * D_ba) + 0.5f : 0.f;
    fca = (D_ca < RCUT) ? 0.5f * __cosf(PI_RC * D_ca) + 0.5f : 0.f;
  }

  float Rbc2 = fmaxf(D_ba * D_ba + D_ca * D_ca - 2.f * D_ba * D_ca * cp, 1e-12f);
  float Rbc  = sqrtf(Rbc2);
  float fbc  = (Rbc < RCUT) ? 0.5f * __cosf(PI_RC * Rbc) + 0.5f : 0.f;
  float cut  = fba * fca * fbc;
  if (cut == 0.f) return;                            // culls the 18-atomic burst

  int sb = ei[ba], sc = ei[ca], da = ei[E + ca];
  int ab = an[sb], ac = an[sc], aa = an[da];
  int off3 = (a2i[aa] * NS + a2i[ab]) * NS + a2i[ac];
  const float* et = G4e + off3 * 3;
  const float* ze = G4z + off3 * 3;
  const float* lm = G4l + off3 * 2;

  float S = Rbc2 + D_ba * D_ba + D_ca * D_ca;
  float A[3];
#pragma unroll
  for (int i = 0; i < 3; ++i) A[i] = __expf(-et[i] * S) * cut;

  // 2^(1-z) * |1+l*c|^z  ==  exp2(1 + z*(log2|1+l*c| - 1))
  float p0 = fabsf(fmaf(lm[0], cp, 1.f));
  float p1 = fabsf(fmaf(lm[1], cp, 1.f));
  float t0 = __log2f(fmaxf(p0, 1e-30f)) - 1.f;
  float t1 = __log2f(fmaxf(p1, 1e-30f)) - 1.f;
  float B[2][3];
#pragma unroll
  for (int z = 0; z < 3; ++z) {
    B[0][z] = exp2f(fmaf(ze[z], t0, 1.f));
    B[1][z] = exp2f(fmaf(ze[z], t1, 1.f));
  }

  int desc4 = imap[(aa * maxa + ab) * maxa + ac];
  float* base = out + (size_t)da * N_COL + 16 + desc4; // col = 16+((e*2+l)*3+z)*3+p
#pragma unroll
  for (int e2 = 0; e2 < 3; ++e2)
#pragma unroll
    for (int l = 0; l < 2; ++l)
#pragma unroll
      for (int z = 0; z < 3; ++z)
        atomic_add_f32(base + ((e2 * 2 + l) * 3 + z) * 3, A[e2] * B[l][z]);
}

// Persistent-grid G4 kernel.  TDM path: double-buffered LDS tiles — wave 0
// issues tile i+1's three DMA descriptors, then s_wait_tensorcnt(3) so only
// the *oldest* three (tile i) must have landed before compute starts.
__global__ void __launch_bounds__(TILE)
acsf_g4(const int* __restrict__ id3_ba, const int* __restrict__ id3_ca,
        const float* __restrict__ cosphi, const int* __restrict__ ei,
        const float* __restrict__ D_st, const int* __restrict__ an,
        const float* __restrict__ G4e, const float* __restrict__ G4z,
        const float* __restrict__ G4l, const int* __restrict__ a2i,
        const int* __restrict__ imap, const float2* __restrict__ etab,
        float* __restrict__ out, int T, int E, int maxa, int ntiles) {
  const int tid = threadIdx.x;
#if USE_TDM
  __shared__ int   sh_ba[2][TILE];
  __shared__ int   sh_ca[2][TILE];
  __shared__ float sh_cp[2][TILE];

  if (tid < 32)                                    // wave 0 drives the TDM
    tdm_issue_tile(id3_ba, id3_ca, cosphi, blockIdx.x, T,
                   &sh_ba[0][0], &sh_ca[0][0], &sh_cp[0][0]);

  int it = 0;
  for (int t = blockIdx.x; t < ntiles; t += gridDim.x, ++it) {
    const int buf   = it & 1;
    const int tnext = t + gridDim.x;
    if (tid < 32) {
      if (tnext < ntiles) {                        // prefetch tile i+1
        tdm_issue_tile(id3_ba, id3_ca, cosphi, tnext, T,
                       &sh_ba[buf ^ 1][0], &sh_ca[buf ^ 1][0],
                       &sh_cp[buf ^ 1][0]);
        __builtin_amdgcn_s_wait_tensorcnt(3);      // tile i complete (in-order)
      } else {
        __builtin_amdgcn_s_wait_tensorcnt(0);      // drain
      }
    }
    __syncthreads();                               // tile i visible to all waves

    int g = t * TILE + tid;
    if (g < T) {
      int   ba = sh_ba[buf][tid];
      int   ca = sh_ca[buf][tid];
      float cp = sh_cp[buf][tid];
      if (ba > ca)                                 // mask (id3_ba > id3_ca)
        g4_body(ba, ca, cp, ei, D_st, an, G4e, G4z, G4l, a2i, imap,
                etab, out, E, maxa);
    }
    __syncthreads();                               // buf reusable for tile i+2
  }
#else
  for (int t = blockIdx.x; t < ntiles; t += gridDim.x) {
    int g = t * TILE + tid;
    if (g < T) {
      int ba = id3_ba[g], ca = id3_ca[g];
      float cp = cosphi[g];
      if (ba > ca)
        g4_body(ba, ca, cp, ei, D_st, an, G4e, G4z, G4l, a2i, imap,
                etab, out, E, maxa);
    }
  }
#endif
}

// ---------------------------------------------------------------------------
extern "C" void kernel_launch(void* const* d_in, const int* in_sizes, int n_in,
                              void* d_out, int out_size, void* d_ws, size_t ws_size,
                              hipStream_t stream) {
  const int*   an     = (const int*)  d_in[0];
  const int*   ei     = (const int*)  d_in[1];
  const float* D_st   = (const float*)d_in[2];
  const int*   id3_ba = (const int*)  d_in[3];
  const int*   id3_ca = (const int*)  d_in[4];
  const float* cosphi = (const float*)d_in[5];
  const float* G2p    = (const float*)d_in[6];
  const float* G4e    = (const float*)d_in[7];
  const float* G4z    = (const float*)d_in[8];
  const float* G4l    = (const float*)d_in[9];
  const int*   a2i    = (const int*)  d_in[10];
  const int*   imap   = (const int*)  d_in[11];
  const int*   imap2  = (const int*)  d_in[12];
  float*       out    = (float*)d_out;

  const int E    = in_sizes[2];
  const int T    = in_sizes[3];
  const int maxa = in_sizes[10];

  float2* etab = (ws_size >= (size_t)E * sizeof(float2)) ? (float2*)d_ws : nullptr;

  // 1) zero the accumulator (harness does NOT re-zero between replays)
  int n4 = (out_size + 3) / 4;
  acsf_zero<<<(n4 + 255) / 256, 256, 0, stream>>>(out, out_size);

  // 2) G2 scatter + per-edge cutoff table
  acsf_g2_edges<<<(E + 255) / 256, 256, 0, stream>>>(ei, D_st, an, G2p, a2i,
                                                     imap2, etab, out, E, maxa);

  // 3) G4 scatter, persistent grid with double-buffered TDM prefetch
  int ntiles = (T + TILE - 1) / TILE;
  int nblk   = ntiles < MAXBLK ? ntiles : MAXBLK;
  acsf_g4<<<nblk, TILE, 0, stream>>>(id3_ba, id3_ca, cosphi, ei, D_st, an,
                                     G4e, G4z, G4l, a2i, imap, etab, out,
                                     T, E, maxa, ntiles);
}